// UncertaintyCalibrator_43224550867324
// MI455X (gfx1250) — compile-verified
//
#include <hip/hip_runtime.h>
#include <math.h>
#include <stdint.h>

// ---------------- problem constants (fixed by the reference) ----------------
constexpr int C_DIM   = 32000;          // classes per row
constexpr int GRID_N  = 100;            // isotonic grid points
constexpr int THREADS = 512;            // 16 wave32 per workgroup
constexpr int WAVES   = THREADS / 32;
constexpr int C4      = C_DIM / 4;      // 16B chunks per row (8000)

// native vector types (HIP's float4 is a class; builtins need real vectors)
typedef float f4 __attribute__((ext_vector_type(4)));
typedef int   i4 __attribute__((ext_vector_type(4)));

typedef __attribute__((address_space(1))) i4* as1_i4p;  // global v4i pointer
typedef __attribute__((address_space(3))) i4* as3_i4p;  // LDS v4i pointer

__device__ __forceinline__ float fast_exp2(float x) {
#if defined(__HIP_DEVICE_COMPILE__) && __has_builtin(__builtin_amdgcn_exp2f)
  return __builtin_amdgcn_exp2f(x);     // v_exp_f32
#else
  return exp2f(x);
#endif
}

__device__ __forceinline__ float wave_max(float v) {
#pragma unroll
  for (int off = 16; off >= 1; off >>= 1)
    v = fmaxf(v, __shfl_xor(v, off, 32));
  return v;
}

__device__ __forceinline__ float wave_sum(float v) {
#pragma unroll
  for (int off = 16; off >= 1; off >>= 1)
    v += __shfl_xor(v, off, 32);
  return v;
}

// One workgroup per row. The whole 125 KB row is staged into LDS once via the
// CDNA5 async global->LDS path (ASYNCcnt-tracked, bypasses VGPRs), then the
// max / sum / emit passes run out of LDS, so HBM traffic is exactly
// 1 read + 1 write of the matrix (roofline-minimal: ~524 MB @ 23.3 TB/s).
__global__ __launch_bounds__(THREADS, 2)
void calib_kernel(const float* __restrict__ logits,
                  const float* __restrict__ logT,
                  const float* __restrict__ iso_x,
                  const float* __restrict__ iso_y,
                  float* __restrict__ out) {
  extern __shared__ __align__(16) float smem[];
  float* row  = smem;                   // [C_DIM]   staged logits
  f4*    row4 = (f4*)row;
  float* mids = smem + C_DIM;           // [99] bin midpoints
  float* ys   = smem + C_DIM + 100;     // [100] isotonic_y
  float* red  = smem + C_DIM + 200;     // [WAVES] reduction scratch

  const int tid  = threadIdx.x;
  const int lane = tid & 31;
  const int wid  = tid >> 5;
  const size_t rowbase = (size_t)blockIdx.x * (size_t)C_DIM;
  const f4* __restrict__ gsrc = (const f4*)(logits + rowbase);
  f4* __restrict__ gdst = (f4*)(out + rowbase);

  // temperature = clamp(exp(logT), 0.1, 10) -- uniform scalar
  float T = expf(logT[0]);
  T = fminf(fmaxf(T, 0.1f), 10.0f);
  const float invT = 1.0f / T;
  const float kk   = invT * 1.4426950408889634f;   // invT * log2(e)

  if (tid < GRID_N)     ys[tid]   = iso_y[tid];
  if (tid < GRID_N - 1) mids[tid] = 0.5f * (iso_x[tid] + iso_x[tid + 1]);

  // ---- phase 0: async-stage the row into LDS (16B per lane per issue) ----
#if defined(__HIP_DEVICE_COMPILE__) && __has_builtin(__builtin_amdgcn_global_load_async_to_lds_b128)
  for (int i = tid; i < C4; i += THREADS) {
    __builtin_amdgcn_global_load_async_to_lds_b128(
        (as1_i4p)(gsrc + i), (as3_i4p)(row4 + i), /*offset=*/0, /*cpol=*/0);
  }
#if __has_builtin(__builtin_amdgcn_s_wait_asynccnt)
  __builtin_amdgcn_s_wait_asynccnt(0);
#else
  asm volatile("s_wait_asynccnt 0x0" ::: "memory");
#endif
#else
  for (int i = tid; i < C4; i += THREADS) row4[i] = gsrc[i];
#endif
  __syncthreads();

  // ---- phase 1: row max (for softmax stability, matches jax.nn.softmax) ----
  float m = -INFINITY;
  for (int i = tid; i < C4; i += THREADS) {
    f4 v = row4[i];
    m = fmaxf(m, fmaxf(fmaxf(v.x, v.y), fmaxf(v.z, v.w)));
  }
  m = wave_max(m);
  if (lane == 0) red[wid] = m;
  __syncthreads();
  float M = red[0];
#pragma unroll
  for (int w = 1; w < WAVES; ++w) M = fmaxf(M, red[w]);
  __syncthreads();   // red is reused below

  // ---- phase 2: softmax denominator ----
  float s = 0.0f;
  for (int i = tid; i < C4; i += THREADS) {
    f4 v = row4[i];
    s += fast_exp2((v.x - M) * kk) + fast_exp2((v.y - M) * kk) +
         fast_exp2((v.z - M) * kk) + fast_exp2((v.w - M) * kk);
  }
  s = wave_sum(s);
  if (lane == 0) red[wid] = s;
  __syncthreads();
  float S = 0.0f;
#pragma unroll
  for (int w = 0; w < WAVES; ++w) S += red[w];
  const float inv_sum = 1.0f / S;

  // ---- phase 3: p -> searchsorted(mids, p, left) -> gather isotonic_y ----
  for (int i = tid; i < C4; i += THREADS) {
    f4 v = row4[i];
    f4 o;
#pragma unroll
    for (int j = 0; j < 4; ++j) {
      float p = fast_exp2((v[j] - M) * kk) * inv_sum;
      // uniform-grid closed form: first i with (i+0.5)/99 >= p
      int idx = (int)ceilf(p * 99.0f - 0.5f);
      idx = idx < 0 ? 0 : (idx > GRID_N - 1 ? GRID_N - 1 : idx);
      // exact fix-up against the true float mids (ties -> lower index)
      while (idx > 0 && mids[idx - 1] >= p) --idx;
      while (idx < GRID_N - 1 && mids[idx] < p) ++idx;
      o[j] = ys[idx];
    }
    __builtin_nontemporal_store(o, gdst + i);   // write-once output, NT hint
  }
}

extern "C" void kernel_launch(void* const* d_in, const int* in_sizes, int n_in,
                              void* d_out, int out_size, void* d_ws, size_t ws_size,
                              hipStream_t stream) {
  (void)n_in; (void)out_size; (void)d_ws; (void)ws_size;
  const float* logits = (const float*)d_in[0];
  const float* logT   = (const float*)d_in[1];
  const float* isox   = (const float*)d_in[2];
  const float* isoy   = (const float*)d_in[3];
  float* out = (float*)d_out;

  const int nrows = in_sizes[0] / C_DIM;                    // 2048
  const size_t smem_bytes = (size_t)(C_DIM + 256) * sizeof(float);  // ~126 KB

  // allow >64KB dynamic LDS (320KB/WGP on CDNA5); ignore result (idempotent)
  (void)hipFuncSetAttribute(reinterpret_cast<const void*>(&calib_kernel),
                            hipFuncAttributeMaxDynamicSharedMemorySize,
                            (int)smem_bytes);

  calib_kernel<<<dim3(nrows), dim3(THREADS), smem_bytes, stream>>>(
      logits, logT, isox, isoy, out);
}